// RN_89919435309107
// MI455X (gfx1250) — compile-verified
//
#include <hip/hip_runtime.h>
#include <hip/hip_bf16.h>

// Relation-network head for MI455X (gfx1250), wave32.
//  - fp32 WMMA (v_wmma_f32_16x16x4_f32) for all GEMMs (exact-precision vs ref)
//  - TDM (tensor_load_to_lds) stages x[n] into LDS in stage 1
//  - no per-lane exec-divergent loads anywhere near the WMMA loops

typedef __attribute__((ext_vector_type(2))) float        v2f;
typedef __attribute__((ext_vector_type(8))) float        v8f;
typedef __attribute__((ext_vector_type(4))) unsigned int v4u;
typedef __attribute__((ext_vector_type(4))) int          v4i;
typedef __attribute__((ext_vector_type(8))) int          v8i;

#define N_OBJ  25
#define N_FEAT 64
#define D0     256

__device__ __forceinline__ v8f wmma_f32_16x16x4(v2f a, v2f b, v8f c) {
    // (neg_a, A, neg_b, B, c_mod, C, reuse_a, reuse_b)
    return __builtin_amdgcn_wmma_f32_16x16x4_f32(false, a, false, b,
                                                 (short)0, c, false, false);
}

// ---------------------------------------------------------------------------
// Tensor Data Mover: copy nElem contiguous f32 from global into LDS.
// D# layout per CDNA5 ISA ch.8: group0 = {flags, lds_addr, global_addr, type},
// group1 = {data_size, tensor/tile dims, strides}. 1-row 2D tile.
// ---------------------------------------------------------------------------
#if __has_builtin(__builtin_amdgcn_tensor_load_to_lds)
  #if __has_include(<hip/amd_detail/amd_gfx1250_TDM.h>)
    #define TDM_ARITY 6   // amdgpu-toolchain / therock headers -> 6-arg builtin
  #else
    #define TDM_ARITY 5   // ROCm 7.2 clang-22 -> 5-arg builtin
  #endif
#endif

#ifdef TDM_ARITY
__device__ __forceinline__ void tdm_load_f32(const float* gsrc, float* lds_dst,
                                             int nElem) {
    const unsigned int lds_off = (unsigned int)(size_t)(void*)lds_dst;
    const unsigned long long ga = (unsigned long long)(size_t)gsrc;
    v4u g0;
    g0[0] = 1u;                                        // count=1 user descriptor
    g0[1] = lds_off;                                   // lds_addr (bytes)
    g0[2] = (unsigned int)(ga & 0xFFFFFFFFu);          // global_addr[31:0]
    g0[3] = (unsigned int)((ga >> 32) & 0x01FFFFFFu)   // global_addr[56:32]
          | (2u << 30);                                // type = 2 ("image")
    v8i g1;
    g1[0] = (int)(2u << 16);                           // data_size = 4 bytes
    g1[1] = (int)(((unsigned)nElem & 0xFFFFu) << 16);  // tensor_dim0[15:0]
    g1[2] = (int)((((unsigned)nElem >> 16) & 0xFFFFu)  // tensor_dim0[31:16]
          | (1u << 16));                               // tensor_dim1 = 1
    g1[3] = (int)(((unsigned)nElem & 0xFFFFu) << 16);  // tile_dim0 = nElem
    g1[4] = 1;                                         // tile_dim1 = 1
    g1[5] = nElem;                                     // tensor_dim0_stride
    g1[6] = 0;
    g1[7] = 0;
    v4i z4 = {0, 0, 0, 0};
  #if TDM_ARITY == 6
    v8i z8 = {0, 0, 0, 0, 0, 0, 0, 0};
    __builtin_amdgcn_tensor_load_to_lds(g0, g1, z4, z4, z8, 0);
  #else
    __builtin_amdgcn_tensor_load_to_lds(g0, g1, z4, z4, 0);
  #endif
}
#endif

// ---------------------------------------------------------------------------
// Stage 1: per-sample fused tap-GEMMs + pairwise relu reduction.
// Grid: 1024 blocks x 256 threads (8 waves).
// LDS: Xs 8 KB + As/Bs 2*26*256*4 = 52 KB  -> 60 KB total.
// ---------------------------------------------------------------------------
__global__ __launch_bounds__(256) void relnet_stage1(
    const float* __restrict__ x,    // (1024, 25*64)
    const float* __restrict__ w0,   // (256, 64) row-major (c, f)
    const float* __restrict__ w1,   // (256, 64)
    const float* __restrict__ cb,   // (256)
    float* __restrict__ h)          // (1024, 256)
{
    __shared__ float Xs[32 * N_FEAT];      // rows 25..31: zero padding
    __shared__ float As[(N_OBJ + 1) * D0]; // row 25 = write-spill row
    __shared__ float Bs[(N_OBJ + 1) * D0];

    const int n     = blockIdx.x;
    const int tid   = threadIdx.x;
    const int wave  = tid >> 5;     // 0..7
    const int lane  = tid & 31;
    const int lhalf = lane >> 4;    // 0/1
    const int l16   = lane & 15;

    const float* xn = x + (size_t)n * (N_OBJ * N_FEAT);

    // Zero the pad rows of Xs (floats 1600..2047).
    for (int idx = N_OBJ * N_FEAT + tid; idx < 32 * N_FEAT; idx += 256)
        Xs[idx] = 0.f;

#ifdef TDM_ARITY
    if (tid < 32) {                         // wave 0 only; TDM ignores EXEC
        tdm_load_f32(xn, Xs, N_OBJ * N_FEAT);
        __builtin_amdgcn_s_wait_tensorcnt(0);
    }
#else
    for (int idx = tid; idx < N_OBJ * N_FEAT; idx += 256)
        Xs[idx] = xn[idx];
#endif
    __syncthreads();

    // 64 wave-tiles: mat(2) x mtile(2) x ntile(16); 8 per wave.
    for (int it = 0; it < 8; ++it) {
        const int T   = wave * 8 + it;
        const int mat = T >> 5;            // 0 -> tap A (w0), 1 -> tap B (w1)
        const int mt  = (T >> 4) & 1;      // object-row tile
        const int nt  = T & 15;            // channel tile
        const float* W = mat ? w1 : w0;

        const int arow = mt * 16 + l16;    // 0..31; rows 25..31 read LDS zeros
        const int ncol = nt * 16 + l16;    // 0..255

        v8f c = {0.f, 0.f, 0.f, 0.f, 0.f, 0.f, 0.f, 0.f};
        #pragma unroll
        for (int k0 = 0; k0 < N_FEAT; k0 += 4) {
            const int ka = k0 + lhalf * 2;
            v2f a, b;
            a[0] = Xs[arow * N_FEAT + ka];      // ds_load_b64, no divergence
            a[1] = Xs[arow * N_FEAT + ka + 1];
            b[0] = W[ncol * N_FEAT + ka];       // global_load_b64
            b[1] = W[ncol * N_FEAT + ka + 1];
            c = wmma_f32_16x16x4(a, b, c);
        }

        float* dst = mat ? Bs : As;
        #pragma unroll
        for (int r = 0; r < 8; ++r) {
            const int m  = mt * 16 + lhalf * 8 + r;   // D layout: M = r + 8*lhalf
            const int ms = (m < N_OBJ) ? m : N_OBJ;   // spill rows 25..31 -> 25
            dst[ms * D0 + ncol] = c[r];
        }
    }
    __syncthreads();

    // Pairwise reduction: thread t owns channel c = t.
    const int ch = tid;
    float a_reg[N_OBJ], b_reg[N_OBJ];
    const float bias = cb[ch];
    #pragma unroll
    for (int i = 0; i < N_OBJ; ++i) {
        a_reg[i] = As[i * D0 + ch] + bias;  // fold conv bias into tap A
        b_reg[i] = Bs[i * D0 + ch];
    }

    float acc = 0.f;
    #pragma unroll
    for (int d = 1; d < N_OBJ; ++d) {
        float part = 0.f;
        #pragma unroll
        for (int i = 0; i + d < N_OBJ; ++i) {
            const float s = a_reg[i] + b_reg[i + d];
            part += fmaxf(s, 0.f);
        }
        acc += part * (1.0f / (float)(N_OBJ - d));   // avg-pool weight, folded
    }
    h[(size_t)n * D0 + ch] = acc * (1.0f / (float)(N_OBJ - 1));
}

// ---------------------------------------------------------------------------
// WMMA GEMM: C = [relu](A @ W^T + bias)
//   A: (M,K) row-major, W: (Nout,K) row-major, C: (M,Nout)
//   M % 16 == 0, K % 4 == 0. NGUARD=false requires Nout % 16 == 0.
// One 16x16 output tile per wave; guards are address clamps (cndmask),
// never exec branches, so EXEC is all-ones across every WMMA.
// ---------------------------------------------------------------------------
template <bool RELU, bool NGUARD>
__global__ __launch_bounds__(256) void gemm_wmma(
    const float* __restrict__ A,
    const float* __restrict__ W,
    const float* __restrict__ bias,
    float* __restrict__ C,
    int M, int K, int Nout)
{
    const int ntn     = (Nout + 15) >> 4;
    const int wave_id = blockIdx.x * (blockDim.x >> 5) + (threadIdx.x >> 5);
    const int total   = (M >> 4) * ntn;
    if (wave_id >= total) return;          // wave-uniform: EXEC stays full

    const int mt    = wave_id / ntn;
    const int nt    = wave_id % ntn;
    const int lane  = threadIdx.x & 31;
    const int lhalf = lane >> 4;
    const int l16   = lane & 15;

    const int arow = mt * 16 + l16;
    const int ncol = nt * 16 + l16;
    // Clamp (not predicate) the weight row: OOB columns compute garbage that
    // is never stored.
    const int ncl  = NGUARD ? (ncol < Nout ? ncol : Nout - 1) : ncol;

    v8f c = {0.f, 0.f, 0.f, 0.f, 0.f, 0.f, 0.f, 0.f};
    #pragma unroll 4
    for (int k0 = 0; k0 < K; k0 += 4) {
        const int ka = k0 + lhalf * 2;
        v2f a, b;
        a[0] = A[(size_t)arow * K + ka];
        a[1] = A[(size_t)arow * K + ka + 1];
        b[0] = W[(size_t)ncl * K + ka];
        b[1] = W[(size_t)ncl * K + ka + 1];
        c = wmma_f32_16x16x4(a, b, c);
    }

    const float bv = bias[ncl];
    #pragma unroll
    for (int r = 0; r < 8; ++r) {
        const int m = mt * 16 + lhalf * 8 + r;
        float v = c[r] + bv;
        if (RELU) v = fmaxf(v, 0.f);
        if (!NGUARD || ncol < Nout)
            C[(size_t)m * Nout + ncol] = v;
    }
}

extern "C" void kernel_launch(void* const* d_in, const int* in_sizes, int n_in,
                              void* d_out, int out_size, void* d_ws, size_t ws_size,
                              hipStream_t stream) {
    const float* x   = (const float*)d_in[0];
    const float* cw0 = (const float*)d_in[1];
    const float* cw1 = (const float*)d_in[2];
    const float* cb  = (const float*)d_in[3];
    const float* w1  = (const float*)d_in[4];
    const float* b1  = (const float*)d_in[5];
    const float* w2  = (const float*)d_in[6];
    const float* b2  = (const float*)d_in[7];
    const float* w3  = (const float*)d_in[8];
    const float* b3  = (const float*)d_in[9];
    float* out = (float*)d_out;              // (1024, 10)

    const int N = 1024;
    float* h1 = (float*)d_ws;                // N x 256
    float* h2 = h1 + (size_t)N * D0;
    float* h3 = h2 + (size_t)N * D0;

    // Stage 1: fused taps + pair reduction
    relnet_stage1<<<N, 256, 0, stream>>>(x, cw0, cw1, cb, h1);

    // MLP layers 1-2: (1024,256)@(256,256)^T -> 1024 waves -> 128 blocks
    {
        int waves  = (N / 16) * (D0 / 16);
        int blocks = (waves * 32 + 255) / 256;
        gemm_wmma<true, false><<<blocks, 256, 0, stream>>>(h1, w1, b1, h2, N, D0, D0);
        gemm_wmma<true, false><<<blocks, 256, 0, stream>>>(h2, w2, b2, h3, N, D0, D0);
    }
    // MLP layer 3: (1024,256)@(256,10)^T -> 64 waves -> 8 blocks
    {
        int waves  = (N / 16) * 1;
        int blocks = (waves * 32 + 255) / 256;
        gemm_wmma<false, true><<<blocks, 256, 0, stream>>>(h3, w3, b3, out, N, D0, 10);
    }
}